// CapsuleSequenceToGraph_3212635537941
// MI455X (gfx1250) — compile-verified
//
#include <hip/hip_runtime.h>
#include <hip/hip_bf16.h>
#include <cmath>

typedef __attribute__((ext_vector_type(2))) float v2f;
typedef __attribute__((ext_vector_type(8))) float v8f;

// ---------------------------------------------------------------------------
// Stage 1: primary capsules.  pri[b,t,n,d] = sum_j x[b,t,j] * W[t,n,j,d]
// One workgroup per t.  X_t [64x64] staged in LDS (row stride 68 to avoid
// bank conflicts on the 16-lane A-fragment column reads).  Each wave runs
// 4 n-blocks of a 64(M) x 16(N) x 64(K) GEMM via V_WMMA_F32_16X16X4_F32.
//
// Fragment layouts (ISA 7.12.2, 32-bit, wave32):
//   A 16x4 : lane L<16 holds A[M=L][K=0](v0), A[L][K=1](v1); lanes 16-31 K=2,3
//   B 4x16 : lane L<16 holds B[K=0][N=L](v0), B[1][L](v1);  lanes 16-31 K=2,3
//   D 16x16: VGPR r = row M=r (lanes 0-15) / M=r+8 (lanes 16-31), N = lane%16
// ---------------------------------------------------------------------------
template <int T>
__global__ __launch_bounds__(256) void prim_caps_kernel(
    const float* __restrict__ X,   // [64, T, 64]
    const float* __restrict__ W,   // [T, 32, 64, 16]
    float* __restrict__ pri)       // [64, T, 32, 16]
{
  __shared__ float Xs[64 * 68];
  const int t   = blockIdx.x;
  const int tid = threadIdx.x;

  // Cooperative load of X_t (64 rows x 64 floats) into LDS.
  {
    const int row  = tid >> 2;
    const int part = (tid & 3) * 16;
    const float* src = X + ((size_t)row * T + t) * 64 + part;
    float*       dst = Xs + row * 68 + part;
#pragma unroll
    for (int q = 0; q < 4; ++q)
      *(float4*)(dst + q * 4) = *(const float4*)(src + q * 4);
  }
  __syncthreads();

  const int wave  = tid >> 5;
  const int lane  = tid & 31;
  const int lhalf = lane >> 4;   // 0: lanes 0-15, 1: lanes 16-31
  const int lmod  = lane & 15;

#pragma unroll
  for (int ni = 0; ni < 4; ++ni) {
    const int n = wave + ni * 8;                 // covers n = 0..31
    // Wn is a [64 j][16 d] row-major block.
    const float* Wn = W + (((size_t)t * 32 + n) * 64) * 16;

    // Preload all 16 B fragments (whole Wn = 4KB per wave, contiguous).
    float b0[16], b1[16];
#pragma unroll
    for (int k = 0; k < 16; ++k) {
      const int j = 4 * k + 2 * lhalf;
      const float* wp = Wn + j * 16 + lmod;
      b0[k] = wp[0];     // row j
      b1[k] = wp[16];    // row j+1
    }

    v8f acc[4] = {};     // 4 M-tiles of 16 rows each (M = 64)
#pragma unroll
    for (int mt = 0; mt < 4; ++mt) {
      const int row = mt * 16 + lmod;
#pragma unroll
      for (int k = 0; k < 16; ++k) {
        const int j = 4 * k + 2 * lhalf;
        const float2 a = *(const float2*)(Xs + row * 68 + j);
        v2f av; av[0] = a.x;   av[1] = a.y;
        v2f bv; bv[0] = b0[k]; bv[1] = b1[k];
        acc[mt] = __builtin_amdgcn_wmma_f32_16x16x4_f32(
            false, av, false, bv, (short)0, acc[mt], false, false);
      }
    }

    // Store D tiles: pri[((b*T + t)*32 + n)*16 + d]
#pragma unroll
    for (int mt = 0; mt < 4; ++mt) {
#pragma unroll
      for (int r = 0; r < 8; ++r) {
        const int brow = mt * 16 + r + 8 * lhalf;
        pri[(((size_t)brow * T + t) * 32 + n) * 16 + lmod] = acc[mt][r];
      }
    }
  }
}

// ---------------------------------------------------------------------------
// Stage 2: dynamic routing (3 iterations + final readout), fused per (b).
// 8 waves; wave w owns t in [w*ROWS, (w+1)*ROWS).  Logits b[t,n] live in
// registers (n == lane, softmax over n == wave32 shuffle butterfly).
// v[n,d] is reduced across waves via LDS each iteration.
// pri re-reads hit the 192MB L2 (<= 67MB per modality).
// ---------------------------------------------------------------------------
template <int T>
__global__ __launch_bounds__(256) void routing_kernel(
    const float* __restrict__ pri,  // [64, T, 32, 16]
    float* __restrict__ out)        // [64, 32, 16]
{
  constexpr int ROWS = T / 8;
  __shared__ float vred[8 * 512];
  __shared__ float vsh[512];

  const int b    = blockIdx.x;
  const int tid  = threadIdx.x;
  const int wave = tid >> 5;
  const int lane = tid & 31;

  float blog[ROWS];
#pragma unroll
  for (int i = 0; i < ROWS; ++i) blog[i] = 0.0f;

  const float* prib = pri + (size_t)b * T * 512;

  for (int iter = 0; iter < 4; ++iter) {
    // ---- phase V: rc = softmax_n(blog); vacc[n=lane,d] += rc * pri ----
    float vacc[16];
#pragma unroll
    for (int d = 0; d < 16; ++d) vacc[d] = 0.0f;

#pragma unroll
    for (int i = 0; i < ROWS; ++i) {
      const int t  = wave * ROWS + i;
      const float bl = blog[i];
      float mx = bl;
#pragma unroll
      for (int off = 16; off > 0; off >>= 1)
        mx = fmaxf(mx, __shfl_xor(mx, off));
      const float e = __expf(bl - mx);
      float s = e;
#pragma unroll
      for (int off = 16; off > 0; off >>= 1)
        s += __shfl_xor(s, off);
      const float rc = e / s;

      const float* p = prib + (size_t)t * 512 + lane * 16;
#pragma unroll
      for (int q = 0; q < 4; ++q) {
        const float4 pv = *(const float4*)(p + q * 4);
        vacc[q * 4 + 0] += rc * pv.x;
        vacc[q * 4 + 1] += rc * pv.y;
        vacc[q * 4 + 2] += rc * pv.z;
        vacc[q * 4 + 3] += rc * pv.w;
      }
    }

#pragma unroll
    for (int d = 0; d < 16; ++d)
      vred[wave * 512 + lane * 16 + d] = vacc[d];
    __syncthreads();

    if (iter < 3) {
      // reduce across waves, apply tanh -> v in LDS
#pragma unroll
      for (int e2 = 0; e2 < 2; ++e2) {
        const int idx = tid * 2 + e2;
        float s = 0.0f;
#pragma unroll
        for (int ww = 0; ww < 8; ++ww) s += vred[ww * 512 + idx];
        vsh[idx] = tanhf(s);
      }
      __syncthreads();

      // ---- phase B: blog[t,n] += sum_d v[n,d] * pri[b,t,n,d] ----
#pragma unroll
      for (int i = 0; i < ROWS; ++i) {
        const int t = wave * ROWS + i;
        const float* p = prib + (size_t)t * 512 + lane * 16;
        const float* v = vsh + lane * 16;
        float dot = 0.0f;
#pragma unroll
        for (int q = 0; q < 4; ++q) {
          const float4 pv = *(const float4*)(p + q * 4);
          const float4 vv = *(const float4*)(v + q * 4);
          dot += pv.x * vv.x + pv.y * vv.y + pv.z * vv.z + pv.w * vv.w;
        }
        blog[i] += dot;
      }
      __syncthreads();   // vred is rewritten next iteration
    } else {
      // final readout: out[b,n,d] = tanh(sum_waves vred)
#pragma unroll
      for (int e2 = 0; e2 < 2; ++e2) {
        const int idx = tid * 2 + e2;
        float s = 0.0f;
#pragma unroll
        for (int ww = 0; ww < 8; ++ww) s += vred[ww * 512 + idx];
        out[(size_t)b * 512 + idx] = tanhf(s);
      }
    }
  }
}

// ---------------------------------------------------------------------------
extern "C" void kernel_launch(void* const* d_in, const int* in_sizes, int n_in,
                              void* d_out, int out_size, void* d_ws, size_t ws_size,
                              hipStream_t stream) {
  (void)in_sizes; (void)n_in; (void)out_size; (void)ws_size;

  const float* x_text   = (const float*)d_in[0];
  const float* x_audio  = (const float*)d_in[1];
  const float* x_video  = (const float*)d_in[2];
  const float* x_frames = (const float*)d_in[3];
  const float* W_t = (const float*)d_in[4];
  const float* W_a = (const float*)d_in[5];
  const float* W_v = (const float*)d_in[6];
  const float* W_f = (const float*)d_in[7];
  float* out = (float*)d_out;
  float* pri = (float*)d_ws;   // reused per modality (max 64*512*512 floats = 64MB)

  // text: T=128
  prim_caps_kernel<128><<<128, 256, 0, stream>>>(x_text, W_t, pri);
  routing_kernel<128><<<64, 256, 0, stream>>>(pri, out + 0);
  // audio: T=512
  prim_caps_kernel<512><<<512, 256, 0, stream>>>(x_audio, W_a, pri);
  routing_kernel<512><<<64, 256, 0, stream>>>(pri, out + 32768);
  // video: T=256
  prim_caps_kernel<256><<<256, 256, 0, stream>>>(x_video, W_v, pri);
  routing_kernel<256><<<64, 256, 0, stream>>>(pri, out + 65536);
  // frames: T=256
  prim_caps_kernel<256><<<256, 256, 0, stream>>>(x_frames, W_f, pri);
  routing_kernel<256><<<64, 256, 0, stream>>>(pri, out + 98304);
}